// drt_9904194585001
// MI455X (gfx1250) — compile-verified
//
#include <hip/hip_runtime.h>

// Pseudo-polar Radon transform via DFT-as-matmul on the CDNA5 fp32 matrix pipe
// (V_WMMA_F32_16X16X4_F32 -- full fp32 precision, required for 512-point DFTs).
//
//   stage1: T   = img · W                 (fft along rows,  norm 1/sqrt(512))
//   stage2: dft = W · T                   (fft along cols   => fft2 ortho)
//   stage3: out = gather(dft) · conj(W)   (linogram gather fused into A load;
//                                          conj(W) == ifft ortho)
//
// Fragment-exact memory layouts (no register shuffles on any load):
//   * Wre_row/Wim_row : row-major split planes  -> stage2 A fragments
//   * Wre_p/Wim_p, T  : paired-row split planes P[(k>>1)*512+col] =
//                       (M[k][col], M[k+1][col]) == WMMA B-fragment layout
//   * dft (D)         : interleaved complex -> stage3 gather gets re+im per b64
//
// All three GEMM stages are software-pipelined one K-group deep: loads for
// group i+1 are issued before the WMMAs of group i, so the s_wait_loadcnt in
// front of each WMMA group covers a full group of latency with compute.

typedef float v2f __attribute__((ext_vector_type(2)));
typedef float v8f __attribute__((ext_vector_type(8)));

#define NN 512
#define MU 768
#define NBATCH 16

__device__ __forceinline__ v8f wmma4(v2f a, v2f b, v8f c) {
  return __builtin_amdgcn_wmma_f32_16x16x4_f32(
      /*neg_a=*/false, a, /*neg_b=*/false, b,
      /*c_mod=*/(short)0, c, /*reuse_a=*/false, /*reuse_b=*/false);
}

__device__ __forceinline__ v2f ldv2(const float* p) { return *(const v2f*)p; }

// ---- twiddles: W[u][x] = exp(-2*pi*i*((u*x) mod 512)/512)/sqrt(512), both layouts
__global__ void k_twiddle(float* __restrict__ Wre_row, float* __restrict__ Wim_row,
                          float* __restrict__ Wre_p,   float* __restrict__ Wim_p) {
  int idx = blockIdx.x * blockDim.x + threadIdx.x;     // 0 .. 512*512-1
  int u = idx >> 9;
  int x = idx & (NN - 1);
  int t = (u * x) & (NN - 1);
  float th = -6.283185307179586f * (float)t / (float)NN;
  float s, c;
  __sincosf(th, &s, &c);
  const float scale = 0.044194173824159216f;           // 1/sqrt(512)
  c *= scale; s *= scale;
  Wre_row[idx] = c;
  Wim_row[idx] = s;
  int pp = 2 * ((u >> 1) * NN + x) + (u & 1);          // paired-row layout
  Wre_p[pp] = c;
  Wim_p[pp] = s;
}

// ---- stage 1: T[b] = img[b] (real) * W ; T written in paired split planes.
__global__ __launch_bounds__(256) void k_stage1(const float* __restrict__ img,
                                                const float* __restrict__ Wre_p,
                                                const float* __restrict__ Wim_p,
                                                float* __restrict__ Tre,
                                                float* __restrict__ Tim) {
  const int lane = threadIdx.x & 31;
  const int wave = threadIdx.x >> 5;
  const int lo = lane & 15;
  const int hi = lane >> 4;
  const int b  = blockIdx.z;
  const int n0 = blockIdx.x * 64;
  const int m0 = blockIdx.y * 128 + wave * 16;

  // loop-invariant bases; per-step offset is k0 (A) / k0*NN + 32*t (B)
  const float* Ap   = img + (size_t)b * NN * NN + (m0 + lo) * NN + 2 * hi;
  const float* BreB = Wre_p + 2 * (hi * NN + n0 + lo);
  const float* BimB = Wim_p + 2 * (hi * NN + n0 + lo);

  v8f cre[4] = {}, cim[4] = {};
  v2f a0, br0[4], bi0[4], a1, br1[4], bi1[4];

  a0 = ldv2(Ap);
#pragma unroll
  for (int t = 0; t < 4; ++t) { br0[t] = ldv2(BreB + 32 * t); bi0[t] = ldv2(BimB + 32 * t); }

  for (int k0 = 0; k0 < NN; k0 += 8) {
    // prefetch group k0+4
    a1 = ldv2(Ap + k0 + 4);
#pragma unroll
    for (int t = 0; t < 4; ++t) {
      br1[t] = ldv2(BreB + (k0 + 4) * NN + 32 * t);
      bi1[t] = ldv2(BimB + (k0 + 4) * NN + 32 * t);
    }
    // compute group k0
#pragma unroll
    for (int t = 0; t < 4; ++t) {
      cre[t] = wmma4(a0, br0[t], cre[t]);
      cim[t] = wmma4(a0, bi0[t], cim[t]);
    }
    // prefetch group k0+8 (wraps to 0 on last iter: valid memory, unused result)
    const int k2 = (k0 + 8) & (NN - 1);
    a0 = ldv2(Ap + k2);
#pragma unroll
    for (int t = 0; t < 4; ++t) {
      br0[t] = ldv2(BreB + k2 * NN + 32 * t);
      bi0[t] = ldv2(BimB + k2 * NN + 32 * t);
    }
    // compute group k0+4
#pragma unroll
    for (int t = 0; t < 4; ++t) {
      cre[t] = wmma4(a1, br1[t], cre[t]);
      cim[t] = wmma4(a1, bi1[t], cim[t]);
    }
  }

  float* TreB = Tre + (size_t)b * NN * NN;
  float* TimB = Tim + (size_t)b * NN * NN;
#pragma unroll
  for (int t = 0; t < 4; ++t)
#pragma unroll
    for (int j = 0; j < 4; ++j) {
      int p   = (m0 >> 1) + j + 4 * hi;      // pair rows (2p, 2p+1), both in-lane
      int col = n0 + t * 16 + lo;
      v2f ore = {cre[t][2 * j], cre[t][2 * j + 1]};
      v2f oim = {cim[t][2 * j], cim[t][2 * j + 1]};
      *(v2f*)(TreB + 2 * (p * NN + col)) = ore;
      *(v2f*)(TimB + 2 * (p * NN + col)) = oim;
    }
}

// ---- stage 2: dft[b] = W * T[b] (complex gemm); D written interleaved complex.
__global__ __launch_bounds__(256) void k_stage2(const float* __restrict__ Tre,
                                                const float* __restrict__ Tim,
                                                const float* __restrict__ Wre_row,
                                                const float* __restrict__ Wim_row,
                                                float* __restrict__ D) {
  const int lane = threadIdx.x & 31;
  const int wave = threadIdx.x >> 5;
  const int lo = lane & 15;
  const int hi = lane >> 4;
  const int b  = blockIdx.z;
  const int n0 = blockIdx.x * 64;
  const int m0 = blockIdx.y * 128 + wave * 16;

  const float* AreB = Wre_row + (m0 + lo) * NN + 2 * hi;
  const float* AimB = Wim_row + (m0 + lo) * NN + 2 * hi;
  const float* BreB = Tre + (size_t)b * NN * NN + 2 * (hi * NN + n0 + lo);
  const float* BimB = Tim + (size_t)b * NN * NN + 2 * (hi * NN + n0 + lo);

  v8f cre[4] = {}, cim[4] = {};
  v2f ar0, ai0, br0[4], bi0[4], ar1, ai1, br1[4], bi1[4];

  ar0 = ldv2(AreB);
  ai0 = ldv2(AimB);
#pragma unroll
  for (int t = 0; t < 4; ++t) { br0[t] = ldv2(BreB + 32 * t); bi0[t] = ldv2(BimB + 32 * t); }

  for (int k0 = 0; k0 < NN; k0 += 8) {
    ar1 = ldv2(AreB + k0 + 4);
    ai1 = ldv2(AimB + k0 + 4);
#pragma unroll
    for (int t = 0; t < 4; ++t) {
      br1[t] = ldv2(BreB + (k0 + 4) * NN + 32 * t);
      bi1[t] = ldv2(BimB + (k0 + 4) * NN + 32 * t);
    }
    {
      v2f ain = -ai0;                        // f32 WMMA has no A/B NEG modifier
#pragma unroll
      for (int t = 0; t < 4; ++t) {
        cre[t] = wmma4(ar0, br0[t], cre[t]);
        cre[t] = wmma4(ain, bi0[t], cre[t]);
        cim[t] = wmma4(ar0, bi0[t], cim[t]);
        cim[t] = wmma4(ai0, br0[t], cim[t]);
      }
    }
    const int k2 = (k0 + 8) & (NN - 1);
    ar0 = ldv2(AreB + k2);
    ai0 = ldv2(AimB + k2);
#pragma unroll
    for (int t = 0; t < 4; ++t) {
      br0[t] = ldv2(BreB + k2 * NN + 32 * t);
      bi0[t] = ldv2(BimB + k2 * NN + 32 * t);
    }
    {
      v2f ain = -ai1;
#pragma unroll
      for (int t = 0; t < 4; ++t) {
        cre[t] = wmma4(ar1, br1[t], cre[t]);
        cre[t] = wmma4(ain, bi1[t], cre[t]);
        cim[t] = wmma4(ar1, bi1[t], cim[t]);
        cim[t] = wmma4(ai1, br1[t], cim[t]);
      }
    }
  }

  float* Db = D + (size_t)b * NN * NN * 2;
#pragma unroll
  for (int t = 0; t < 4; ++t)
#pragma unroll
    for (int r = 0; r < 8; ++r) {
      int row = m0 + r + 8 * hi;
      int col = n0 + t * 16 + lo;
      v2f o = {cre[t][r], cim[t][r]};        // interleaved for stage-3 gather
      *(v2f*)(Db + 2 * (row * NN + col)) = o;
    }
}

// ---- stage 3: out[b] = gather(dft[b]) * conj(W)
// Linogram (n=512, p=2): m<512 : (row,col) = (k, k*m mod 512)
//                        m>=512: (row,col) = (2*k*(m-512) mod 512, k)
__device__ __forceinline__ void gather2(const float* __restrict__ Db, int m, int k_a,
                                        v2f& s0, v2f& s1) {
  int k_b = k_a + 1;
  int r0 = (m < NN) ? k_a : ((2 * k_a * (m - NN)) & (NN - 1));
  int c0 = (m < NN) ? ((k_a * m) & (NN - 1)) : k_a;
  int r1 = (m < NN) ? k_b : ((2 * k_b * (m - NN)) & (NN - 1));
  int c1 = (m < NN) ? ((k_b * m) & (NN - 1)) : k_b;
  s0 = ldv2(Db + 2 * (r0 * NN + c0));
  s1 = ldv2(Db + 2 * (r1 * NN + c1));
}

__global__ __launch_bounds__(256) void k_stage3(const float* __restrict__ D,
                                                const float* __restrict__ Wre_p,
                                                const float* __restrict__ Wim_p,
                                                float* __restrict__ out) {
  const int lane = threadIdx.x & 31;
  const int wave = threadIdx.x >> 5;
  const int lo = lane & 15;
  const int hi = lane >> 4;
  const int b  = blockIdx.z;
  const int n0 = blockIdx.x * 64;
  const int m0 = blockIdx.y * 128 + wave * 16;         // up to 768

  const float* Db   = D + (size_t)b * NN * NN * 2;
  const float* BreB = Wre_p + 2 * (hi * NN + n0 + lo);
  const float* BimB = Wim_p + 2 * (hi * NN + n0 + lo);
  const int m = m0 + lo;            // 16-aligned tiles; m<512 is wave-uniform

  v8f cre[4] = {}, cim[4] = {};
  v2f s0a, s1a, wr0[4], wi0[4], s0b, s1b, wr1[4], wi1[4];

  gather2(Db, m, 2 * hi, s0a, s1a);
#pragma unroll
  for (int t = 0; t < 4; ++t) { wr0[t] = ldv2(BreB + 32 * t); wi0[t] = ldv2(BimB + 32 * t); }

  for (int k0 = 0; k0 < NN; k0 += 8) {
    gather2(Db, m, k0 + 4 + 2 * hi, s0b, s1b);
#pragma unroll
    for (int t = 0; t < 4; ++t) {
      wr1[t] = ldv2(BreB + (k0 + 4) * NN + 32 * t);
      wi1[t] = ldv2(BimB + (k0 + 4) * NN + 32 * t);
    }
    {
      v2f are = {s0a[0], s1a[0]};
      v2f aim = {s0a[1], s1a[1]};
      v2f arn = -are;
      // S * conj(W): re = Sre*Wre + Sim*Wim ; im = Sim*Wre - Sre*Wim
#pragma unroll
      for (int t = 0; t < 4; ++t) {
        cre[t] = wmma4(are, wr0[t], cre[t]);
        cre[t] = wmma4(aim, wi0[t], cre[t]);
        cim[t] = wmma4(aim, wr0[t], cim[t]);
        cim[t] = wmma4(arn, wi0[t], cim[t]);
      }
    }
    const int k2 = (k0 + 8) & (NN - 1);
    gather2(Db, m, k2 + 2 * hi, s0a, s1a);
#pragma unroll
    for (int t = 0; t < 4; ++t) {
      wr0[t] = ldv2(BreB + k2 * NN + 32 * t);
      wi0[t] = ldv2(BimB + k2 * NN + 32 * t);
    }
    {
      v2f are = {s0b[0], s1b[0]};
      v2f aim = {s0b[1], s1b[1]};
      v2f arn = -are;
#pragma unroll
      for (int t = 0; t < 4; ++t) {
        cre[t] = wmma4(are, wr1[t], cre[t]);
        cre[t] = wmma4(aim, wi1[t], cre[t]);
        cim[t] = wmma4(aim, wr1[t], cim[t]);
        cim[t] = wmma4(arn, wi1[t], cim[t]);
      }
    }
  }

  float* Ob = out + (size_t)b * MU * NN * 2;
#pragma unroll
  for (int t = 0; t < 4; ++t)
#pragma unroll
    for (int r = 0; r < 8; ++r) {
      int row = m0 + r + 8 * hi;
      int col = n0 + t * 16 + lo;
      v2f o = {cre[t][r], cim[t][r]};   // (re,im) interleave == stack([..],-1)
      *(v2f*)(Ob + 2 * (row * NN + col)) = o;
    }
}

extern "C" void kernel_launch(void* const* d_in, const int* in_sizes, int n_in,
                              void* d_out, int out_size, void* d_ws, size_t ws_size,
                              hipStream_t stream) {
  (void)in_sizes; (void)n_in; (void)out_size; (void)ws_size;
  const float* img = (const float*)d_in[0];
  float* out = (float*)d_out;

  // Workspace (68 MB):
  //   Wre_row/Wim_row : 1 MB each   (row-major planes, stage2 A)
  //   Wre_p/Wim_p     : 1 MB each   (paired-row planes, stage1/3 B)
  //   Tre/Tim         : 16 MB each  (paired-row planes)
  //   D               : 32 MB       (interleaved complex)
  float* Wre_row = (float*)d_ws;
  float* Wim_row = Wre_row + (size_t)NN * NN;
  float* Wre_p   = Wim_row + (size_t)NN * NN;
  float* Wim_p   = Wre_p   + (size_t)NN * NN;
  float* Tre     = Wim_p   + (size_t)NN * NN;
  float* Tim     = Tre     + (size_t)NBATCH * NN * NN;
  float* Dm      = Tim     + (size_t)NBATCH * NN * NN;

  k_twiddle<<<(NN * NN) / 256, 256, 0, stream>>>(Wre_row, Wim_row, Wre_p, Wim_p);

  dim3 blk(256);
  dim3 g12(NN / 64, NN / 128, NBATCH);   // 8 x 4 x 16
  k_stage1<<<g12, blk, 0, stream>>>(img, Wre_p, Wim_p, Tre, Tim);
  k_stage2<<<g12, blk, 0, stream>>>(Tre, Tim, Wre_row, Wim_row, Dm);

  dim3 g3(NN / 64, MU / 128, NBATCH);    // 8 x 6 x 16
  k_stage3<<<g3, blk, 0, stream>>>(Dm, Wre_p, Wim_p, out);
}